// Dopri5Block_25005299598014
// MI455X (gfx1250) — compile-verified
//
#include <hip/hip_runtime.h>
#include <hip/hip_bf16.h>
#include <math.h>

// ---------------------------------------------------------------------------
// DoPri5(4) integrator for f(t,y) = tanh(y@W + b + t), y: 4096x512 fp32.
// GEMMs run on V_WMMA_F32_16X16X4_F32 (CDNA5 wave32 WMMA). B is staged into
// LDS pre-swizzled into per-lane float2 fragments so each WMMA operand is a
// single aligned ds_load_b64 (no VGPR repack movs). All tensors stay L2
// resident (~73MB < 192MB); step control is on-device so the 12-step scan is
// one capturable graph.
// ---------------------------------------------------------------------------

typedef float v2f __attribute__((ext_vector_type(2)));
typedef float v8f __attribute__((ext_vector_type(8)));

#define DM   4096            // rows of y
#define DD   512             // feature dim (K and N)
#define NELEM (DM * DD)      // 2,097,152

#define BM 64
#define BN 128
#define KC 32
#define NKQ (KC / 4)         // 8 K-quads per chunk
#define BPAD 144             // float2 stride for Bp "col" dim (+32-bank skew)

#define RTOL 1e-3f
#define ATOL 1e-6f
#define SAFETY 0.9f
#define H_MIN 1e-3f
#define H_MAX 1e30f
#define T_END 1.0f

// scalar state layout in ws (floats):
// sc[0]=t  sc[1]=h  sc[2]=h_eff  sc[4]=ok flag  sc[5]=done flag
// partials at ws+64 (1024 entries), big buffers from ws+4096.

// ------------------------------- init --------------------------------------
__global__ __launch_bounds__(256) void init_kernel(const float* __restrict__ x,
                                                   float* __restrict__ y,
                                                   float* __restrict__ sc) {
  int i = blockIdx.x * 256 + threadIdx.x;
  y[i] = x[i];
  if (blockIdx.x == 0 && threadIdx.x == 0) {
    sc[0] = 0.0f;   // t
    sc[1] = 1.0f;   // h_init = max(H_MIN, min(H_MAX, min(H0, STEPS*H0))) = 1
    sc[2] = 0.0f;
    sc[3] = 0.0f;
    sc[4] = 0.0f;
    sc[5] = 0.0f;
  }
}

// ----------------------------- pre-step ------------------------------------
__global__ void pre_step_kernel(float* __restrict__ sc) {
  float t = sc[0];
  float h = sc[1];
  int done = (t >= T_END - 1e-7f);
  float heff = fminf(h, T_END - t);
  heff = fmaxf(heff, 0.0f);
  sc[2] = heff;
  sc[5] = done ? 1.0f : 0.0f;
}

// --------------------- fused stage: yi build + GEMM + tanh -----------------
// k_i = tanh( (y + heff * sum_j a_j * k_j) @ W + b + (t + c_i*heff) )
__global__ __launch_bounds__(256)
void stage_kernel(const float* __restrict__ y,
                  const float* __restrict__ W,
                  const float* __restrict__ bias,
                  const float* __restrict__ sc,
                  const float* __restrict__ ka0, float a0,
                  const float* __restrict__ ka1, float a1,
                  const float* __restrict__ ka2, float a2,
                  const float* __restrict__ ka3, float a3,
                  const float* __restrict__ ka4, float a4,
                  const float* __restrict__ ka5, float a5,
                  float ci,
                  float* __restrict__ kout) {
  __shared__ float As[BM][KC + 2];        // 64 x 34 floats
  // Pre-swizzled B: Bp[kq][hi][col] = { W[k0+4kq+hi][col], W[k0+4kq+hi+2][col] }
  // hi == (lane>>4); per-lane fragment is one aligned 64-bit LDS load.
  __shared__ v2f Bp[NKQ][2][BPAD];        // 8 x 2 x 144 float2

  const int tid  = threadIdx.x;
  const int lane = tid & 31;
  const int wave = tid >> 5;              // 0..7
  const int wm   = wave & 1;              // 32-row group
  const int wn   = wave >> 1;             // 32-col group (0..3)
  const int m0   = blockIdx.x * BM;
  const int n0   = blockIdx.y * BN;

  const int lhi  = lane >> 4;             // 0 or 1
  const int llo  = lane & 15;

  const float heff = sc[2];
  const float tval = sc[0] + ci * heff;

  v8f acc[2][2];                          // [row tile][col tile], 32x32 patch
#pragma unroll
  for (int rt = 0; rt < 2; ++rt)
#pragma unroll
    for (int ct = 0; ct < 2; ++ct) { v8f z = {}; acc[rt][ct] = z; }

  for (int k0 = 0; k0 < DD; k0 += KC) {
    // prefetch next W tile into caches (global_prefetch_b8)
    if (k0 + KC < DD) {
      __builtin_prefetch(&W[(size_t)(k0 + KC) * DD + n0 + (tid & 127)], 0, 0);
    }
    // stage A tile with on-the-fly stage combination
#pragma unroll
    for (int e = tid; e < BM * KC; e += 256) {
      int r = e >> 5;          // / KC
      int c = e & (KC - 1);
      int gi = (m0 + r) * DD + (k0 + c);
      float v = y[gi];
      float s = 0.0f;
      if (a0 != 0.0f) s += a0 * ka0[gi];
      if (a1 != 0.0f) s += a1 * ka1[gi];
      if (a2 != 0.0f) s += a2 * ka2[gi];
      if (a3 != 0.0f) s += a3 * ka3[gi];
      if (a4 != 0.0f) s += a4 * ka4[gi];
      if (a5 != 0.0f) s += a5 * ka5[gi];
      As[r][c] = v + heff * s;
    }
    // stage B (W) tile, pre-swizzled into per-lane fragments
#pragma unroll
    for (int e = tid; e < NKQ * 2 * BN; e += 256) {
      int col = e & (BN - 1);
      int idx = e >> 7;        // 0..15
      int kq  = idx >> 1;
      int hi  = idx & 1;
      int krow = k0 + 4 * kq + hi;
      v2f t2;
      t2.x = W[(size_t)krow * DD + (n0 + col)];
      t2.y = W[(size_t)(krow + 2) * DD + (n0 + col)];
      Bp[kq][hi][col] = t2;
    }
    __syncthreads();

#pragma unroll
    for (int kq = 0; kq < NKQ; ++kq) {
      int kb = (kq << 2) + (lhi << 1);
      v2f af[2];
#pragma unroll
      for (int rt = 0; rt < 2; ++rt) {
        int arow = 32 * wm + 16 * rt + llo;
        af[rt] = *(const v2f*)&As[arow][kb];      // aligned ds_load_b64
      }
      v2f bf[2];
#pragma unroll
      for (int ct = 0; ct < 2; ++ct) {
        bf[ct] = Bp[kq][lhi][32 * wn + 16 * ct + llo];  // aligned ds_load_b64
      }
#pragma unroll
      for (int rt = 0; rt < 2; ++rt)
#pragma unroll
        for (int ct = 0; ct < 2; ++ct) {
          acc[rt][ct] = __builtin_amdgcn_wmma_f32_16x16x4_f32(
              false, af[rt], false, bf[ct], (short)0, acc[rt][ct], false,
              false);
        }
    }
    __syncthreads();
  }

  // epilogue: bias + t + tanh, scatter per C/D layout
#pragma unroll
  for (int rt = 0; rt < 2; ++rt) {
#pragma unroll
    for (int ct = 0; ct < 2; ++ct) {
      int col = n0 + 32 * wn + 16 * ct + llo;
      float bb = bias[col];
      int rbase = m0 + 32 * wm + 16 * rt + (lhi << 3);
#pragma unroll
      for (int r = 0; r < 8; ++r) {
        float v = acc[rt][ct][r] + bb + tval;
        kout[(size_t)(rbase + r) * DD + col] = tanhf(v);
      }
    }
  }
}

// ------------------------- finalize: y5/y4/err -----------------------------
__global__ __launch_bounds__(256)
void finalize_kernel(const float* __restrict__ y,
                     const float* __restrict__ k0, const float* __restrict__ k1,
                     const float* __restrict__ k2, const float* __restrict__ k3,
                     const float* __restrict__ k4, const float* __restrict__ k5,
                     const float* __restrict__ k6,
                     const float* __restrict__ sc,
                     float* __restrict__ y5buf,
                     float* __restrict__ partials) {
  (void)k1; // B5[1] == B4[1] == 0
  const float heff = sc[2];
  // B5 (order 5) nonzero coefficients
  const float b50 = 35.0f / 384.0f, b52 = 500.0f / 1113.0f,
              b53 = 125.0f / 192.0f, b54 = -2187.0f / 6784.0f,
              b55 = 11.0f / 84.0f;
  // B4 (order 4) nonzero coefficients
  const float c40 = 5179.0f / 57600.0f, c42 = 7571.0f / 16695.0f,
              c43 = 393.0f / 640.0f, c44 = -92097.0f / 339200.0f,
              c45 = 187.0f / 2100.0f, c46 = 1.0f / 40.0f;

  float sum = 0.0f;
#pragma unroll
  for (int it = 0; it < 8; ++it) {
    int i = blockIdx.x * 2048 + it * 256 + threadIdx.x;
    float yv = y[i];
    float v0 = k0[i], v2 = k2[i], v3 = k3[i], v4 = k4[i], v5 = k5[i], v6 = k6[i];
    float y5 = yv + heff * (b50 * v0 + b52 * v2 + b53 * v3 + b54 * v4 + b55 * v5);
    float y4 = yv + heff * (c40 * v0 + c42 * v2 + c43 * v3 + c44 * v4 +
                            c45 * v5 + c46 * v6);
    float scale = ATOL + RTOL * fmaxf(fabsf(y5), fabsf(y4));
    float d = (y5 - y4) / scale;
    sum += d * d;
    y5buf[i] = y5;
  }
  __shared__ float sh[256];
  sh[threadIdx.x] = sum;
  __syncthreads();
  for (int s = 128; s > 0; s >>= 1) {
    if (threadIdx.x < s) sh[threadIdx.x] += sh[threadIdx.x + s];
    __syncthreads();
  }
  if (threadIdx.x == 0) partials[blockIdx.x] = sh[0];
}

// -------------------- deterministic reduce + step control ------------------
__global__ __launch_bounds__(256)
void reduce_control_kernel(const float* __restrict__ partials, int nparts,
                           float* __restrict__ sc) {
  __shared__ float sh[256];
  float s = 0.0f;
  for (int i = threadIdx.x; i < nparts; i += 256) s += partials[i];
  sh[threadIdx.x] = s;
  __syncthreads();
  for (int st = 128; st > 0; st >>= 1) {
    if (threadIdx.x < st) sh[threadIdx.x] += sh[threadIdx.x + st];
    __syncthreads();
  }
  if (threadIdx.x == 0) {
    float err = sqrtf(sh[0] / (float)NELEM);
    float t = sc[0], h = sc[1], heff = sc[2];
    bool done = (sc[5] != 0.0f);
    bool accept = (err <= 1.0f) || (heff <= H_MIN * 1.0001f);
    bool ok = (!done) && accept;
    sc[4] = ok ? 1.0f : 0.0f;
    if (ok) sc[0] = t + heff;
    float factor = SAFETY * powf(err + 1e-12f, -0.2f);
    factor = fminf(fmaxf(factor, 0.2f), 5.0f);
    float hn = fminf(fmaxf(heff * factor, H_MIN), H_MAX);
    sc[1] = done ? h : hn;
  }
}

// ------------------------- conditional commit ------------------------------
__global__ __launch_bounds__(256)
void commit_kernel(const float* __restrict__ sc, float* __restrict__ y,
                   const float* __restrict__ y5) {
  if (sc[4] != 0.0f) {
#pragma unroll
    for (int it = 0; it < 4; ++it) {
      int i = blockIdx.x * 1024 + it * 256 + threadIdx.x;
      y[i] = y5[i];
    }
  }
}

// ------------------------------ copy-out -----------------------------------
__global__ __launch_bounds__(256)
void copyout_kernel(const float* __restrict__ y, float* __restrict__ out) {
  int i = blockIdx.x * 256 + threadIdx.x;
  out[i] = y[i];
}

// ---------------------------------------------------------------------------
extern "C" void kernel_launch(void* const* d_in, const int* in_sizes, int n_in,
                              void* d_out, int out_size, void* d_ws,
                              size_t ws_size, hipStream_t stream) {
  (void)in_sizes; (void)n_in; (void)out_size; (void)ws_size;
  const float* x = (const float*)d_in[0];   // 4096x512
  const float* W = (const float*)d_in[1];   // 512x512
  const float* b = (const float*)d_in[2];   // 512
  float* out = (float*)d_out;

  float* ws = (float*)d_ws;
  float* sc = ws;                 // 64 scalar floats
  float* partials = ws + 64;      // 1024 used
  float* base = ws + 4096;
  float* y = base;
  float* kb[7];
  for (int i = 0; i < 7; ++i) kb[i] = base + (size_t)(i + 1) * NELEM;
  float* y5 = base + (size_t)8 * NELEM;

  // DoPri tableau (A rows, C nodes)
  static const double Atab[7][6] = {
      {0, 0, 0, 0, 0, 0},
      {1.0 / 5.0, 0, 0, 0, 0, 0},
      {3.0 / 40.0, 9.0 / 40.0, 0, 0, 0, 0},
      {44.0 / 45.0, -56.0 / 15.0, 32.0 / 9.0, 0, 0, 0},
      {19372.0 / 6561.0, -25360.0 / 2187.0, 64448.0 / 6561.0, -212.0 / 729.0, 0, 0},
      {9017.0 / 3168.0, -355.0 / 33.0, 46732.0 / 5247.0, 49.0 / 176.0,
       -5103.0 / 18656.0, 0},
      {35.0 / 384.0, 0.0, 500.0 / 1113.0, 125.0 / 192.0, -2187.0 / 6784.0,
       11.0 / 84.0}};
  static const float Ctab[7] = {0.0f, 0.2f, 0.3f, 0.8f, 8.0f / 9.0f, 1.0f, 1.0f};

  dim3 sgrid(DM / BM, DD / BN);   // 64 x 4 blocks

  init_kernel<<<NELEM / 256, 256, 0, stream>>>(x, y, sc);

  for (int step = 0; step < 12; ++step) {
    pre_step_kernel<<<1, 1, 0, stream>>>(sc);
    for (int i = 0; i < 7; ++i) {
      const float* kp[6];
      float av[6];
      for (int j = 0; j < 6; ++j) {
        float aj = (j < i) ? (float)Atab[i][j] : 0.0f;
        av[j] = aj;
        kp[j] = (aj != 0.0f) ? kb[j] : y;  // dummy (never read when a==0)
      }
      stage_kernel<<<sgrid, 256, 0, stream>>>(
          y, W, b, sc, kp[0], av[0], kp[1], av[1], kp[2], av[2], kp[3], av[3],
          kp[4], av[4], kp[5], av[5], Ctab[i], kb[i]);
    }
    finalize_kernel<<<NELEM / 2048, 256, 0, stream>>>(
        y, kb[0], kb[1], kb[2], kb[3], kb[4], kb[5], kb[6], sc, y5, partials);
    reduce_control_kernel<<<1, 256, 0, stream>>>(partials, NELEM / 2048, sc);
    commit_kernel<<<NELEM / 1024, 256, 0, stream>>>(sc, y, y5);
  }

  copyout_kernel<<<NELEM / 256, 256, 0, stream>>>(y, out);
}